// E55ScalarNoProj_47656957116866
// MI455X (gfx1250) — compile-verified
//
#include <hip/hip_runtime.h>
#include <cstdint>
#include <cstddef>

// CDNA5 / gfx1250. wave32. fp32 WMMA 16x16x4 turns the intra-chunk linear
// recurrence into dense matrix ops (chunked-scan formulation).
//
// Roofline: x = 134 MB fits in the 192 MB L2. Pass 1 pulls x into L2 (RT
// policy); pass 3 re-reads it with NT (last use) and writes out with NT so
// the 134 MB of output doesn't evict x. HBM traffic ~= 268 MB => ~11.5 us
// at 23.3 TB/s.
//
// Kernels are templated on D so row strides become compile-time constants:
// all intra-chunk row offsets (<= 61 KB) fold into the 24-bit signed
// instruction offset of global_load/store, killing the v_add_nc_u64 chains.

typedef __attribute__((ext_vector_type(2))) float v2f;
typedef __attribute__((ext_vector_type(8))) float v8f;

#define LSEG  256   // timesteps per segment (2^8; see the squaring in pass 2)
#define CHUNK 16    // timesteps per WMMA chunk

// Fast sigmoid: v_exp_f32 + v_rcp_f32 (~1 ulp), avoids the IEEE div sequence
// (v_div_scale/v_div_fmas/v_div_fixup ~10 VALU ops) the compiler otherwise emits.
__device__ __forceinline__ float sigmoidf_(float v) {
    return __builtin_amdgcn_rcpf(1.0f + __expf(-v));
}

// ---------------------------------------------------------------------------
// Pass 1: per-segment end-state assuming h_in = 0.
//   summary[b][s][d] = sum_{j=0..L-1} lam^(L-1-j) * u_j ,  u = lam*silu(x)+bias
// Grid: B * (D/256) * S blocks, 256 threads (one channel each). Coalesced 1KB/row.
// x is read with default (RT) policy on purpose: warms the 192MB L2 for pass 3.
// ---------------------------------------------------------------------------
template <int DC>   // DC > 0: compile-time D; DC == 0: runtime D
__global__ __launch_bounds__(256) void seg_summary_kernel(
    const float* __restrict__ x, const float* __restrict__ loglam,
    const float* __restrict__ bias, float* __restrict__ summary,
    int B, int T, int Drt, int S)
{
    const int D = (DC > 0) ? DC : Drt;

    int nbd = D >> 8;
    int bi  = blockIdx.x;
    int s   = bi % S;  bi /= S;
    int db  = bi % nbd;
    int b   = bi / nbd;
    int d   = (db << 8) + threadIdx.x;

    const float lam = sigmoidf_(loglam[0]);
    const float bch = bias[d];
    const float* xp = x + ((size_t)b * T + (size_t)s * LSEG) * D + d;

    float acc = 0.0f;
#pragma unroll 8
    for (int i = 0; i < LSEG; ++i) {
        float xv = xp[(size_t)i * D];
        float u  = lam * (xv * sigmoidf_(xv)) + bch;
        acc = lam * acc + u;
    }
    __builtin_nontemporal_store(acc, &summary[((size_t)b * S + s) * D + d]);
}

// ---------------------------------------------------------------------------
// Pass 2: scan over segments per channel.
//   carry_{s+1} = lam^LSEG * carry_s + summary[s];  hstart[s] = carry_s
// Also emits the h_final output (second element of the reference tuple).
// ---------------------------------------------------------------------------
__global__ __launch_bounds__(256) void seg_scan_kernel(
    const float* __restrict__ loglam, const float* __restrict__ h0,
    const float* __restrict__ summary, float* __restrict__ hstart,
    float* __restrict__ hfinal, int B, int D, int S)
{
    int idx = blockIdx.x * 256 + threadIdx.x;   // b*D + d, grid sized exactly
    int b = idx / D, d = idx % D;

    const float lam = sigmoidf_(loglam[0]);
    float lamL = lam;
#pragma unroll
    for (int i = 0; i < 8; ++i) lamL *= lamL;   // lam^256 == lam^LSEG

    float carry = h0[idx];
    for (int s = 0; s < S; ++s) {
        size_t o = ((size_t)b * S + s) * D + d;
        hstart[o] = carry;
        carry = lamL * carry + summary[o];
    }
    hfinal[idx] = carry;
}

// ---------------------------------------------------------------------------
// Pass 3: WMMA chunked scan. One wave owns (batch b, 16-channel tile, segment).
// Per 16-step chunk:  H[16t x 16d] = L @ U + lam^{i+1} * h_start
// via 4x V_WMMA_F32_16X16X4_F32 (K-slices of the 16x16 lower-triangular L).
//
// fp32 operand layouts (ISA 7.12.2):
//   A 16x4:  lanes 0-15 -> M=lane, v0=K0,v1=K1 ; lanes 16-31 -> M=lane-16, v0=K2,v1=K3
//   B 4x16:  lanes 0-15 -> N=lane, v0=K0,v1=K1 ; lanes 16-31 -> N=lane-16, v0=K2,v1=K3
//   C/D:     vgpr r, lanes 0-15 -> M=r, N=lane ; lanes 16-31 -> M=r+8, N=lane-16
// Carry (row 15 lives in vgpr7 of the upper half) is broadcast with
// ds_swizzle_b32 SWAPX16 (offset 0x401F).
// ---------------------------------------------------------------------------
template <int DC>   // DC > 0: compile-time D; DC == 0: runtime D
__global__ __launch_bounds__(256) void scan_wmma_kernel(
    const float* __restrict__ x, const float* __restrict__ loglam,
    const float* __restrict__ bias, const float* __restrict__ hstart,
    float* __restrict__ out, int B, int T, int Drt, int S)
{
    const int D = (DC > 0) ? DC : Drt;

    const int lane = threadIdx.x & 31;
    const int wv   = threadIdx.x >> 5;
    const int wid  = blockIdx.x * 8 + wv;        // 8 waves / block
    const int dtiles = D >> 4;

    int dt = wid % dtiles;
    int r2 = wid / dtiles;
    int s  = r2 % S;
    int b  = r2 / S;

    const int n    = lane & 15;                  // M (A) / N (B,C,D) index
    const int half = lane >> 4;                  // K-half selector
    const int ch   = (dt << 4) + n;              // this lane's channel

    const float lam = sigmoidf_(loglam[0]);

    // A slices: amat[k][vi] = L[n][4k + 2*half + vi], L[m][c] = (m>=c)?lam^(m-c):0
    float amat[4][2];
#pragma unroll
    for (int k = 0; k < 4; ++k) { amat[k][0] = 0.0f; amat[k][1] = 0.0f; }
    {
        float p = 1.0f;
#pragma unroll
        for (int e = 0; e < 16; ++e) {           // static indices only -> no spills
#pragma unroll
            for (int k = 0; k < 4; ++k)
#pragma unroll
                for (int vi = 0; vi < 2; ++vi)
                    if (n - (4 * k + 2 * half + vi) == e) amat[k][vi] = p;
            p *= lam;
        }
    }

    // Carry-in coefficients: cvec[r] = lam^(r + 8*half + 1)
    float cvec[8];
    {
        float l2 = lam * lam, l4 = l2 * l2, l8 = l4 * l4;
        float p = lam * (half ? l8 : 1.0f);
#pragma unroll
        for (int r = 0; r < 8; ++r) { cvec[r] = p; p *= lam; }
    }

    const float bch = bias[ch];
    float hs = hstart[((size_t)b * S + s) * D + ch];

    // Per-lane bases; with constexpr D every row offset below is an
    // instruction-immediate (max 15*D*4 = 61440 bytes < 2^23).
    const float* xrow = x   + ((size_t)b * T + (size_t)s * LSEG) * D
                           + (size_t)(2 * half) * D + ch;   // first K row for this lane
    float*       orow = out + ((size_t)b * T + (size_t)s * LSEG) * D
                           + (size_t)(8 * half) * D + ch;   // first C row for this lane

    const v2f a0 = { amat[0][0], amat[0][1] };
    const v2f a1 = { amat[1][0], amat[1][1] };
    const v2f a2 = { amat[2][0], amat[2][1] };
    const v2f a3 = { amat[3][0], amat[3][1] };

    for (int c = 0; c < LSEG / CHUNK; ++c) {
        // Build B operands: u at rows xrow + (4k+vi)*D (k,vi compile-time).
        // NT loads: this is the last read of x; no need to re-cache it.
        v2f bm[4];
#pragma unroll
        for (int k = 0; k < 4; ++k) {
#pragma unroll
            for (int vi = 0; vi < 2; ++vi) {
                float xv = __builtin_nontemporal_load(&xrow[(size_t)(4 * k + vi) * D]);
                bm[k][vi] = lam * (xv * sigmoidf_(xv)) + bch;
            }
        }

        // C := lam^{row+1} * h_start  (carry-in rank-1 term)
        v8f acc;
#pragma unroll
        for (int r = 0; r < 8; ++r) acc[r] = cvec[r] * hs;

        // H = L @ U + C  via 4 chained fp32 WMMAs (K=4 slices)
        acc = __builtin_amdgcn_wmma_f32_16x16x4_f32(false, a0, false, bm[0],
                                                    (short)0, acc, false, false);
        acc = __builtin_amdgcn_wmma_f32_16x16x4_f32(false, a1, false, bm[1],
                                                    (short)0, acc, false, false);
        acc = __builtin_amdgcn_wmma_f32_16x16x4_f32(false, a2, false, bm[2],
                                                    (short)0, acc, false, false);
        acc = __builtin_amdgcn_wmma_f32_16x16x4_f32(false, a3, false, bm[3],
                                                    (short)0, acc, false, false);

        // New carry = H[15, n] -> vgpr7 of upper half; swap halves to lower.
        float last = acc[7];
        int sw = __builtin_amdgcn_ds_swizzle(__float_as_int(last), 0x401F); // SWAPX16
        hs = half ? last : __int_as_float(sw);

        // Self-gated outputs: h * silu(h). NT stores: out is never re-read,
        // keep it from evicting x in L2. Row offsets r*D are immediates.
#pragma unroll
        for (int r = 0; r < 8; ++r) {
            float h = acc[r];
            float o = h * h * sigmoidf_(h);
            __builtin_nontemporal_store(o, &orow[(size_t)r * D]);
        }

        xrow += (size_t)CHUNK * D;   // single pointer bump per chunk
        orow += (size_t)CHUNK * D;
    }
}

// ---------------------------------------------------------------------------
extern "C" void kernel_launch(void* const* d_in, const int* in_sizes, int n_in,
                              void* d_out, int out_size, void* d_ws, size_t ws_size,
                              hipStream_t stream) {
    const float* x      = (const float*)d_in[0];   // [B,T,D]
    const float* h0     = (const float*)d_in[1];   // [B,D]
    const float* loglam = (const float*)d_in[2];   // [1]
    const float* bias   = (const float*)d_in[3];   // [D]
    float* out = (float*)d_out;                    // [B,T,D] ++ [B,D]

    const int D  = in_sizes[3];
    const int BD = in_sizes[1];
    const int B  = BD / D;
    const int T  = in_sizes[0] / BD;
    const int S  = T / LSEG;                       // segments per sequence

    // Workspace: summary[B][S][D], hstart[B][S][D]  (2 * 512 KB for these shapes)
    float* summary = (float*)d_ws;
    float* hstartp = summary + (size_t)B * S * D;
    float* hfinal  = out + (size_t)B * T * D;

    dim3 blk(256);
    int g1 = B * (D >> 8) * S;                     // 512 blocks
    int g2 = (B * D) >> 8;                         // 32 blocks
    int waves = B * S * (D >> 4);                  // 8192 waves
    int g3 = waves >> 3;                           // 1024 blocks of 8 waves

    if (D == 1024) {
        seg_summary_kernel<1024><<<g1, blk, 0, stream>>>(x, loglam, bias, summary, B, T, D, S);
        seg_scan_kernel<<<g2, blk, 0, stream>>>(loglam, h0, summary, hstartp, hfinal, B, D, S);
        scan_wmma_kernel<1024><<<g3, blk, 0, stream>>>(x, loglam, bias, hstartp, out, B, T, D, S);
    } else {
        seg_summary_kernel<0><<<g1, blk, 0, stream>>>(x, loglam, bias, summary, B, T, D, S);
        seg_scan_kernel<<<g2, blk, 0, stream>>>(loglam, h0, summary, hstartp, hfinal, B, D, S);
        scan_wmma_kernel<0><<<g3, blk, 0, stream>>>(x, loglam, bias, hstartp, out, B, T, D, S);
    }
}